// GNN_55705725829411
// MI455X (gfx1250) — compile-verified
//
#include <hip/hip_runtime.h>

typedef __attribute__((ext_vector_type(2))) float v2f;
typedef __attribute__((ext_vector_type(8))) float v8f;

#define N_NODES 50000
#define N_EDGES 500000
#define HID 128
#define LDSPITCH (HID + 4)   // +16B row pad: lane bank stride 4 -> conflict-free ds_load_b64

// ---------------- utility: zero a float buffer (grid-stride, float4) ----------------
__global__ void zero_f32(float* __restrict__ p, long n) {
  long i = (long)blockIdx.x * blockDim.x + threadIdx.x;
  long stride = (long)gridDim.x * blockDim.x;
  long n4 = n >> 2;  // all our buffers are multiple-of-4 sized
  float4* p4 = (float4*)p;
  for (long j = i; j < n4; j += stride) p4[j] = make_float4(0.f, 0.f, 0.f, 0.f);
}

// ---------------- degree: deg[dst[e]] += 1 ----------------
__global__ void degree_kernel(const int* __restrict__ dst, float* __restrict__ deg, int E) {
  int e = blockIdx.x * blockDim.x + threadIdx.x;
  if (e < E) atomicAdd(&deg[dst[e]], 1.0f);
}

// ---------------- invdeg = 1 / max(deg, 1) ----------------
__global__ void invdeg_kernel(float* __restrict__ deg, int n) {
  int i = blockIdx.x * blockDim.x + threadIdx.x;
  if (i < n) deg[i] = 1.0f / fmaxf(deg[i], 1.0f);
}

// ---------------- scatter: agg[dst[e]][:] += x[src[e]][:], one wave32 per edge ----------------
// lane l handles floats [4l, 4l+4): one b128 load + 4 f32 atomics (no-return -> STOREcnt).
__global__ void scatter_kernel(const float* __restrict__ x,
                               const int* __restrict__ src,
                               const int* __restrict__ dst,
                               float* __restrict__ agg, int E) {
  int wave = (int)(((long)blockIdx.x * blockDim.x + threadIdx.x) >> 5);
  int lane = threadIdx.x & 31;
  if (wave >= E) return;
  int s = src[wave];
  int d = dst[wave];
  const float4 v = *(const float4*)(x + (long)s * HID + lane * 4);
  float* o = agg + (long)d * HID + lane * 4;
  atomicAdd(o + 0, v.x);
  atomicAdd(o + 1, v.y);
  atomicAdd(o + 2, v.z);
  atomicAdd(o + 3, v.w);
}

// ---------------- fused SAGE layer GEMM ----------------
// out[r][:] = act( (agg[r]*invdeg[r]) @ Wl + x[r] @ Wr + b )
// block = 32*(OUTD/16) threads; each wave owns one 16-wide column tile.
// B fragments (weights) are register-resident for the whole block; the block
// sweeps STRIPS row-strips of 16, so the inner loop is ds_load + v_wmma only.
// The two fused GEMMs use independent accumulator chains (accL/accR) so
// alternating WMMAs have no RAW dependency on each other.
template <int OUTD, bool RELU, int STRIPS>
__global__ __launch_bounds__(32 * (OUTD / 16)) void sage_gemm(
    const float* __restrict__ xin,     // [N, 128]
    const float* __restrict__ agg,     // [N, 128] summed messages
    const float* __restrict__ invdeg,  // [N]
    const float* __restrict__ Wl,      // [128, OUTD] row-major
    const float* __restrict__ Wr,      // [128, OUTD] row-major
    const float* __restrict__ bias,    // [OUTD]
    float* __restrict__ out,           // [N, OUTD]
    int n_nodes) {
  __shared__ float sX[16][LDSPITCH];  // self features
  __shared__ float sA[16][LDSPITCH];  // mean-aggregated neighbor features

  const int tid = threadIdx.x;
  const int nthreads = 32 * (OUTD / 16);
  const int wave = tid >> 5;
  const int lane = tid & 31;
  const int half = lane >> 4;  // 0: K+{0,1}, rows 0-7 ; 1: K+{2,3}, rows 8-15
  const int r = lane & 15;     // A: row M ; B/D: col N
  const int col = wave * 16 + r;

  // ---- hoist B fragments (this wave's 16 W-columns, all K) into registers ----
  v2f BL[HID / 4], BR[HID / 4];
#pragma unroll
  for (int j = 0; j < HID / 4; ++j) {
    const int ka = 4 * j + 2 * half;
    BL[j][0] = Wl[(long)ka * OUTD + col];
    BL[j][1] = Wl[(long)(ka + 1) * OUTD + col];
    BR[j][0] = Wr[(long)ka * OUTD + col];
    BR[j][1] = Wr[(long)(ka + 1) * OUTD + col];
  }
  const float bn = bias[col];

  for (int s = 0; s < STRIPS; ++s) {
    const int row0 = (blockIdx.x * STRIPS + s) * 16;
    if (row0 >= n_nodes) break;  // n_nodes % 16 == 0 -> strips are never partial

    // ---- stage 16x128 strip of x and mean-agg into LDS (float4) ----
    for (int i = tid; i < 16 * (HID / 4); i += nthreads) {
      const int rr = i / (HID / 4);
      const int c4 = (i % (HID / 4)) * 4;
      const long base = (long)(row0 + rr) * HID + c4;
      float4 xv = *(const float4*)(xin + base);
      float4 av = *(const float4*)(agg + base);
      const float iv = invdeg[row0 + rr];
      av.x *= iv; av.y *= iv; av.z *= iv; av.w *= iv;
      *(float4*)&sX[rr][c4] = xv;
      *(float4*)&sA[rr][c4] = av;
    }
    __syncthreads();

    // ---- K=128: two independent 32-deep WMMA chains (no cross dependency) ----
    v8f accL = {};
    v8f accR = {};
#pragma unroll
    for (int j = 0; j < HID / 4; ++j) {
      const int ka = 4 * j + 2 * half;
      v2f aL, aR;
      aL[0] = sA[r][ka];
      aL[1] = sA[r][ka + 1];
      aR[0] = sX[r][ka];
      aR[1] = sX[r][ka + 1];
      accL = __builtin_amdgcn_wmma_f32_16x16x4_f32(false, aL, false, BL[j],
                                                   (short)0, accL, false, false);
      accR = __builtin_amdgcn_wmma_f32_16x16x4_f32(false, aR, false, BR[j],
                                                   (short)0, accR, false, false);
    }

    // ---- epilogue: combine chains, bias (+ relu), write per 16x16 f32 C/D layout ----
#pragma unroll
    for (int i = 0; i < 8; ++i) {
      const int gr = row0 + half * 8 + i;  // VGPR i -> row 8*half+i, col = lane%16
      float v = accL[i] + accR[i] + bn;
      if (RELU) v = fmaxf(v, 0.0f);
      out[(long)gr * OUTD + col] = v;
    }
    __syncthreads();  // protect LDS reuse next strip
  }
}

extern "C" void kernel_launch(void* const* d_in, const int* in_sizes, int n_in,
                              void* d_out, int out_size, void* d_ws, size_t ws_size,
                              hipStream_t stream) {
  const float* x   = (const float*)d_in[0];
  const int* eidx  = (const int*)d_in[1];  // [2, E]: row0 = src, row1 = dst
  const float* W1l = (const float*)d_in[2];
  const float* W1r = (const float*)d_in[3];
  const float* b1  = (const float*)d_in[4];
  const float* W2l = (const float*)d_in[5];
  const float* W2r = (const float*)d_in[6];
  const float* b2  = (const float*)d_in[7];
  const float* W3l = (const float*)d_in[8];
  const float* W3r = (const float*)d_in[9];
  const float* b3  = (const float*)d_in[10];
  float* out = (float*)d_out;

  const int* src = eidx;
  const int* dst = eidx + N_EDGES;

  // workspace layout
  float* deg = (float*)d_ws;               // N  (becomes invdeg)
  float* agg = deg + N_NODES;              // N*128 (reused per layer)
  float* h1  = agg + (long)N_NODES * HID;  // N*128
  float* h2  = h1 + (long)N_NODES * HID;   // N*128

  const int ZB = 2048;
  const int eBlocks = (N_EDGES + 255) / 256;
  const int scatBlocks = (int)(((long)N_EDGES * 32 + 255) / 256);
  constexpr int STRIPS = 4;  // 64 rows per block: 4x weight-fetch amortization, 782 blocks
  const int gemmBlocks = (N_NODES + 16 * STRIPS - 1) / (16 * STRIPS);

  // degree -> invdeg (edge structure is layer-invariant)
  zero_f32<<<ZB, 256, 0, stream>>>(deg, N_NODES);
  degree_kernel<<<eBlocks, 256, 0, stream>>>(dst, deg, N_EDGES);
  invdeg_kernel<<<(N_NODES + 255) / 256, 256, 0, stream>>>(deg, N_NODES);

  // layer 1
  zero_f32<<<ZB, 256, 0, stream>>>(agg, (long)N_NODES * HID);
  scatter_kernel<<<scatBlocks, 256, 0, stream>>>(x, src, dst, agg, N_EDGES);
  sage_gemm<128, true, STRIPS><<<gemmBlocks, 256, 0, stream>>>(x, agg, deg, W1l, W1r, b1, h1, N_NODES);

  // layer 2
  zero_f32<<<ZB, 256, 0, stream>>>(agg, (long)N_NODES * HID);
  scatter_kernel<<<scatBlocks, 256, 0, stream>>>(h1, src, dst, agg, N_EDGES);
  sage_gemm<128, true, STRIPS><<<gemmBlocks, 256, 0, stream>>>(h1, agg, deg, W2l, W2r, b2, h2, N_NODES);

  // layer 3 (OUT=64, no relu) -> d_out
  zero_f32<<<ZB, 256, 0, stream>>>(agg, (long)N_NODES * HID);
  scatter_kernel<<<scatBlocks, 256, 0, stream>>>(h2, src, dst, agg, N_EDGES);
  sage_gemm<64, false, STRIPS><<<gemmBlocks, 128, 0, stream>>>(h2, agg, deg, W3l, W3r, b3, out, N_NODES);
}